// BilinearPooling_4698694222672
// MI455X (gfx1250) — compile-verified
//
#include <hip/hip_runtime.h>
#include <hip/hip_bf16.h>

typedef __attribute__((ext_vector_type(2))) float v2f;
typedef __attribute__((ext_vector_type(4))) float v4f;
typedef __attribute__((ext_vector_type(8))) float v8f;

#define B_SZ  32
#define D_IN  1024
#define D_OUT 2048
#define EPS_N 1e-12f

// ---------------------------------------------------------------------------
// Kernel 1: P = X * W^T via V_WMMA_F32_16X16X4_F32 (one wave per 16x16 tile).
//   X: [32, 1024] row-major, W: [2048, 1024] row-major (so B = W^T).
//   blockIdx.x = N tile (128), blockIdx.y = M tile (2), blockIdx.z = {x,y}.
// A frag (16x4 f32):  lane L -> row m_base+(L&15), K pair = (L>>4)*2.
// B frag (4x16 f32):  lane L -> col n_base+(L&15), K pair = (L>>4)*2
//                     == contiguous read of W row n along K.
// C/D (16x16 f32):    VGPR r -> M = r + 8*(L>>4), N = L&15.
// ---------------------------------------------------------------------------
__global__ __launch_bounds__(32) void proj_wmma(const float* __restrict__ x,
                                                const float* __restrict__ y,
                                                const float* __restrict__ W,
                                                float* __restrict__ xp,
                                                float* __restrict__ yp) {
    const int nTile = blockIdx.x;      // 0..127
    const int mTile = blockIdx.y;      // 0..1
    const int which = blockIdx.z;      // 0: x->xp, 1: y->yp
    const float* __restrict__ A   = which ? y  : x;
    float*       __restrict__ out = which ? yp : xp;

    const int lane  = threadIdx.x;     // 0..31 (wave32)
    const int row16 = lane & 15;
    const int khalf = lane >> 4;       // 0 or 1

    const float* aRow = A + (size_t)(mTile * 16 + row16) * D_IN + khalf * 2;
    const float* wRow = W + (size_t)(nTile * 16 + row16) * D_IN + khalf * 2;

    v8f c = {};
    for (int k = 0; k < D_IN; k += 4) {
        v2f a = *(const v2f*)(aRow + k);
        v2f b = *(const v2f*)(wRow + k);
        // 8 args: (neg_a, A, neg_b, B, c_mod, C, reuse_a, reuse_b)
        c = __builtin_amdgcn_wmma_f32_16x16x4_f32(false, a, false, b,
                                                  (short)0, c, false, false);
    }

    const int nOut = nTile * 16 + row16;
#pragma unroll
    for (int r = 0; r < 8; ++r) {
        const int mOut = mTile * 16 + r + 8 * khalf;
        out[(size_t)mOut * D_OUT + nOut] = c[r];
    }
}

// ---------------------------------------------------------------------------
// Kernel 2: per-row sum of squares over the 64 projected rows (xp then yp,
// contiguous in workspace). One 256-thread block per row.
// ---------------------------------------------------------------------------
__global__ __launch_bounds__(256) void rownorm(const float* __restrict__ p,
                                               float* __restrict__ ssq) {
    __shared__ float red[8];
    const int row = blockIdx.x;        // 0..63
    const float* r = p + (size_t)row * D_OUT;

    float s = 0.0f;
    for (int j = threadIdx.x; j < D_OUT; j += 256) {
        const float v = r[j];
        s += v * v;
    }
    // wave32 butterfly reduce
    for (int off = 16; off > 0; off >>= 1)
        s += __shfl_down(s, off, 32);
    if ((threadIdx.x & 31) == 0) red[threadIdx.x >> 5] = s;
    __syncthreads();
    if (threadIdx.x == 0) {
        float t = 0.0f;
#pragma unroll
        for (int w = 0; w < 8; ++w) t += red[w];
        ssq[row] = t;
    }
}

// ---------------------------------------------------------------------------
// Kernel 3: out[b,i,j] = xp[b,i] * yp[b,j] / max(||xp_b||*||yp_b||, eps).
// Pure 512 MB streaming store -> float4 non-temporal (bypass L2; 23.3 TB/s
// is the ceiling). Each thread: 1 x b128 NT store.
//   grid = (D_OUT*D_OUT / 1024, B) ; block = 256 ; 2 blocks per output row i.
// ---------------------------------------------------------------------------
__global__ __launch_bounds__(256) void outer_store(const float* __restrict__ xp,
                                                   const float* __restrict__ yp,
                                                   const float* __restrict__ ssq,
                                                   float* __restrict__ out) {
    const int b   = blockIdx.y;
    const int lin = blockIdx.x;                  // 0..4095
    const int i     = lin >> 1;                  // output row within sample
    const int jBase = (lin & 1) * 1024 + threadIdx.x * 4;

    const float norm = sqrtf(ssq[b]) * sqrtf(ssq[B_SZ + b]);
    const float inv  = 1.0f / fmaxf(norm, EPS_N);
    const float xv   = xp[(size_t)b * D_OUT + i] * inv;

    const v4f yv = *(const v4f*)(yp + (size_t)b * D_OUT + jBase);
    v4f o;
    o.x = xv * yv.x; o.y = xv * yv.y; o.z = xv * yv.z; o.w = xv * yv.w;

    v4f* dst = (v4f*)(out + (size_t)b * (size_t)D_OUT * D_OUT
                          + (size_t)i * D_OUT + jBase);
    __builtin_nontemporal_store(o, dst);
}

// ---------------------------------------------------------------------------
extern "C" void kernel_launch(void* const* d_in, const int* in_sizes, int n_in,
                              void* d_out, int out_size, void* d_ws, size_t ws_size,
                              hipStream_t stream) {
    const float* x = (const float*)d_in[0];   // [32, 1024]
    const float* y = (const float*)d_in[1];   // [32, 1024]
    const float* W = (const float*)d_in[2];   // [2048, 1024]
    float* out = (float*)d_out;               // [32, 2048*2048]

    float* xp  = (float*)d_ws;                // [32, 2048]
    float* yp  = xp + (size_t)B_SZ * D_OUT;   // [32, 2048] (contiguous after xp)
    float* ssq = yp + (size_t)B_SZ * D_OUT;   // [64]

    // 1) projections (WMMA f32)
    proj_wmma<<<dim3(D_OUT / 16, B_SZ / 16, 2), 32, 0, stream>>>(x, y, W, xp, yp);
    // 2) row sum-of-squares for xp and yp (64 rows total, contiguous)
    rownorm<<<dim3(2 * B_SZ), 256, 0, stream>>>(xp, ssq);
    // 3) normalized outer product, streaming NT stores (bandwidth-bound)
    outer_store<<<dim3((D_OUT * D_OUT) / 1024, B_SZ), 256, 0, stream>>>(xp, yp, ssq, out);
}